// EgdeConv_13915694039584
// MI455X (gfx1250) — compile-verified
//
#include <hip/hip_runtime.h>
#include <hip/hip_bf16.h>

// EdgeConv on MI455X (gfx1250): fp32 WMMA 16x16x4, two-pass with recompute
// (recompute ap_res/ue_res instead of spilling 128MB to HBM; 23.3 TB/s makes
// bytes ~10x more expensive than the extra 8.6 GFLOP of f32 WMMA work).

#define A_N 256
#define U_N 1024
#define D_N 64
#define K2  128          // 2*D
#define WAVES 4          // 4 waves * 32 lanes = 128 threads/block
#define TLD 132          // LDS row stride in floats: 16B-aligned (528B), bank-spread

typedef float v2f __attribute__((ext_vector_type(2)));
typedef float v8f __attribute__((ext_vector_type(8)));

// ---- One 16(rows) x 64(cols) GEMM with K=128 from an LDS tile. -------------
// A fragment (16x4 f32): lane m=l&15 supplies row m; VGPR pair holds
// K = kb + 2*(l>>4) + {0,1}.  B fragment (4x16): lane n=l&15 supplies col n,
// same K-pair selection.  Wt is stored n-major ([64][128]) so B is one v2f load.
__device__ __forceinline__ void mm128(const float* t, const float* __restrict__ Wt,
                                      v8f c[4], int lane) {
  const int mn = lane & 15;
  const int kh = (lane >> 4) * 2;
  for (int kb = 0; kb < K2; kb += 4) {
    const int k0 = kb + kh;
    v2f a = *(const v2f*)(t + mn * TLD + k0);
#pragma unroll
    for (int tt = 0; tt < 4; ++tt) {
      v2f b = *(const v2f*)(Wt + (size_t)(tt * 16 + mn) * K2 + k0);
      c[tt] = __builtin_amdgcn_wmma_f32_16x16x4_f32(
          false, a, false, b, (short)0, c[tt], false, false);
    }
  }
}

__device__ __forceinline__ void bias_act(v8f c[4], const float* __restrict__ b,
                                         int lane, bool relu) {
  const int n = lane & 15;
#pragma unroll
  for (int tt = 0; tt < 4; ++tt) {
    const float bv = b[tt * 16 + n];
#pragma unroll
    for (int j = 0; j < 8; ++j) {
      float x = c[tt][j] + bv;
      c[tt][j] = relu ? fmaxf(x, 0.f) : x;
    }
  }
}

// Load 16 rows x 64 floats (row stride in floats) into tile cols [colOff,colOff+64)
__device__ __forceinline__ void load_tile_rows(float* t, const float* __restrict__ src,
                                               int rowStride, int lane, int colOff) {
#pragma unroll
  for (int i = 0; i < 8; ++i) {
    const int flat = i * 32 + lane;
    const int r = flat >> 4, c4 = (flat & 15) * 4;
    float4 v = *(const float4*)(src + (size_t)r * rowStride + c4);
    *(float4*)(t + r * TLD + colOff + c4) = v;
  }
}

// Broadcast one 64-float vector into cols 0..63 of all 16 rows
__device__ __forceinline__ void fill_bcast(float* t, const float* __restrict__ h, int lane) {
  const float v0 = h[lane];
  const float v1 = h[lane + 32];
#pragma unroll
  for (int r = 0; r < 16; ++r) {
    t[r * TLD + lane]      = v0;
    t[r * TLD + lane + 32] = v1;
  }
}

// ---- Prep: transpose weights (n-major), swap W3 K-halves, zero accumulators.
__global__ void k_prep(const float* __restrict__ W1, const float* __restrict__ W2,
                       const float* __restrict__ W3, float* __restrict__ ws) {
  const int idx = blockIdx.x * blockDim.x + threadIdx.x;
  if (idx < 8192) {                              // W1t[n][k] = W1[k][n]
    const int nn = idx >> 7, k = idx & 127;
    ws[idx] = W1[k * 64 + nn];
  } else if (idx < 16384) {                      // W2t
    const int i = idx - 8192, nn = i >> 7, k = i & 127;
    ws[idx] = W2[k * 64 + nn];
  } else if (idx < 24576) {                      // W3t with halves swapped:
    const int i = idx - 16384, nn = i >> 7, k = i & 127;
    ws[idx] = W3[(size_t)((k + 64) & 127) * 64 + nn];   // tile = [agg | edge]
  } else if (idx < 106496) {                     // apSum (16K) + ueSum (64K) = 0
    ws[idx] = 0.0f;
  }
}

// ---- Pass 1a: ap_sum[a] = sum_u relu([ap_hid[a]|edge[a,u]] @ W1 + b1) ------
__global__ __launch_bounds__(128) void k_ap_sum(const float* __restrict__ ap,
                                                const float* __restrict__ edge,
                                                const float* __restrict__ W1t,
                                                const float* __restrict__ b1,
                                                float* __restrict__ apSum) {
  __shared__ float sm[WAVES][16][TLD];
  const int lane = threadIdx.x & 31, w = threadIdx.x >> 5;
  const int a  = blockIdx.x >> 4;                 // 16 blocks per AP (U/64)
  const int u0 = (blockIdx.x & 15) * 64 + w * 16;
  float* t = &sm[w][0][0];

  fill_bcast(t, ap + (size_t)a * 64, lane);                       // cols 0..63
  load_tile_rows(t, edge + ((size_t)a * U_N + u0) * 64, 64, lane, 64);

  v8f c[4] = {};
  mm128(t, W1t, c, lane);
  bias_act(c, b1, lane, true);

  const int n = lane & 15;
#pragma unroll
  for (int tt = 0; tt < 4; ++tt) {
    float s = 0.f;
#pragma unroll
    for (int j = 0; j < 8; ++j) s += c[tt][j];   // rows j+8*(l>>4)
    s += __shfl_xor(s, 16, 32);                  // combine row halves
    if (lane < 16) atomicAdd(&apSum[(size_t)a * 64 + tt * 16 + n], s);
  }
}

// ---- Pass 1b: ue_sum[u] = sum_a relu([ue_hid[u]|edge[a,u]] @ W2 + b2) ------
__global__ __launch_bounds__(128) void k_ue_sum(const float* __restrict__ ue,
                                                const float* __restrict__ edge,
                                                const float* __restrict__ W2t,
                                                const float* __restrict__ b2,
                                                float* __restrict__ ueSum) {
  __shared__ float sm[WAVES][16][TLD];
  const int lane = threadIdx.x & 31, w = threadIdx.x >> 5;
  const int u  = blockIdx.x >> 2;                 // 4 blocks per UE (A/64)
  const int a0 = (blockIdx.x & 3) * 64 + w * 16;
  float* t = &sm[w][0][0];

  fill_bcast(t, ue + (size_t)u * 64, lane);
  load_tile_rows(t, edge + ((size_t)a0 * U_N + u) * 64, U_N * 64, lane, 64);

  v8f c[4] = {};
  mm128(t, W2t, c, lane);
  bias_act(c, b2, lane, true);

  const int n = lane & 15;
#pragma unroll
  for (int tt = 0; tt < 4; ++tt) {
    float s = 0.f;
#pragma unroll
    for (int j = 0; j < 8; ++j) s += c[tt][j];
    s += __shfl_xor(s, 16, 32);
    if (lane < 16) atomicAdd(&ueSum[(size_t)u * 64 + tt * 16 + n], s);
  }
}

// ---- Pass 2: recompute ap_res/ue_res, form agg, final GEMM, write output ---
__global__ __launch_bounds__(128) void k_edge_out(
    const float* __restrict__ ap, const float* __restrict__ ue,
    const float* __restrict__ edge,
    const float* __restrict__ W1t, const float* __restrict__ b1,
    const float* __restrict__ W2t, const float* __restrict__ b2,
    const float* __restrict__ W3t, const float* __restrict__ b3,
    const float* __restrict__ apSum, const float* __restrict__ ueSum,
    float* __restrict__ out) {
  __shared__ float sm[WAVES][16][TLD];
  const int lane = threadIdx.x & 31, w = threadIdx.x >> 5;
  const int a  = blockIdx.x >> 4;
  const int u0 = (blockIdx.x & 15) * 64 + w * 16;
  float* t = &sm[w][0][0];

  load_tile_rows(t, edge + ((size_t)a * U_N + u0) * 64, 64, lane, 64); // cols 64..127
  fill_bcast(t, ap + (size_t)a * 64, lane);                            // cols 0..63

  v8f ac[4] = {};
  mm128(t, W1t, ac, lane);
  bias_act(ac, b1, lane, true);                  // ap_res fragment

  load_tile_rows(t, ue + (size_t)u0 * 64, 64, lane, 0);  // ue_hid rows -> cols 0..63
  v8f uc[4] = {};
  mm128(t, W2t, uc, lane);
  bias_act(uc, b2, lane, true);                  // ue_res fragment

  // agg = apSum[a] - ap_res + ueSum[u] - ue_res ; stash in LDS cols 0..63
  const int n = lane & 15, kh = lane >> 4;
#pragma unroll
  for (int tt = 0; tt < 4; ++tt) {
    const float as = apSum[(size_t)a * 64 + tt * 16 + n];
#pragma unroll
    for (int j = 0; j < 8; ++j) {
      const int m = j + 8 * kh;
      const float us = ueSum[(size_t)(u0 + m) * 64 + tt * 16 + n];
      t[m * TLD + tt * 16 + n] = as - ac[tt][j] + us - uc[tt][j];
    }
  }

  // final GEMM: tile = [agg | edge], W3t already has K-halves swapped
  v8f oc[4] = {};
  mm128(t, W3t, oc, lane);
  bias_act(oc, b3, lane, false);

#pragma unroll
  for (int tt = 0; tt < 4; ++tt) {
#pragma unroll
    for (int j = 0; j < 8; ++j) {
      const int m = j + 8 * kh;
      out[((size_t)a * U_N + u0 + m) * 64 + tt * 16 + n] = oc[tt][j];
    }
  }
}

extern "C" void kernel_launch(void* const* d_in, const int* in_sizes, int n_in,
                              void* d_out, int out_size, void* d_ws, size_t ws_size,
                              hipStream_t stream) {
  const float* ap   = (const float*)d_in[0];
  const float* ue   = (const float*)d_in[1];
  const float* edge = (const float*)d_in[2];
  const float* W1   = (const float*)d_in[3];
  const float* b1   = (const float*)d_in[4];
  const float* W2   = (const float*)d_in[5];
  const float* b2   = (const float*)d_in[6];
  const float* W3   = (const float*)d_in[7];
  const float* b3   = (const float*)d_in[8];
  float* out = (float*)d_out;
  float* ws  = (float*)d_ws;

  float* W1t   = ws;            // 8192 floats
  float* W2t   = ws + 8192;     // 8192
  float* W3t   = ws + 16384;    // 8192
  float* apSum = ws + 24576;    // 256*64
  float* ueSum = ws + 40960;    // 1024*64   (total 416 KB)

  k_prep<<<(106496 + 255) / 256, 256, 0, stream>>>(W1, W2, W3, ws);
  k_ap_sum<<<A_N * (U_N / 64), 128, 0, stream>>>(ap, edge, W1t, b1, apSum);
  k_ue_sum<<<U_N * (A_N / 64), 128, 0, stream>>>(ue, edge, W2t, b2, ueSum);
  k_edge_out<<<A_N * (U_N / 64), 128, 0, stream>>>(ap, ue, edge, W1t, b1, W2t, b2,
                                                   W3t, b3, apSum, ueSum, out);
}